// BasicBlock8_4947802325431
// MI455X (gfx1250) — compile-verified
//
#include <hip/hip_runtime.h>

// MI455X / gfx1250, wave32. Implicit-GEMM convs via V_WMMA_F32_16X16X4_F32.
// A fragments: packed channel-pairs -> one global_load_b64 per fragment.
// B fragments: pixel-major LDS tile (pitch 34) -> one ds_load_b64 per fragment.

typedef __attribute__((ext_vector_type(2))) float v2f;
typedef __attribute__((ext_vector_type(8))) float v8f;

#define CIN 256
#define BHW 1024      // 32*32 pixels per image

__device__ __forceinline__ v8f wmma4(v2f a, v2f b, v8f c) {
  // (neg_a, A, neg_b, B, c_mod, C, reuse_a, reuse_b)
  return __builtin_amdgcn_wmma_f32_16x16x4_f32(false, a, false, b, (short)0, c,
                                               false, false);
}

// ---------------------------------------------------------------- prep kernels

// Fold BN: scale = g/sqrt(v+eps), shift = b - m*scale
__global__ void bnprep_kernel(const float* __restrict__ g, const float* __restrict__ b,
                              const float* __restrict__ m, const float* __restrict__ v,
                              float* __restrict__ scale, float* __restrict__ shift) {
  int i = threadIdx.x;
  float s = g[i] * rsqrtf(v[i] + 1e-5f);
  scale[i] = s;
  shift[i] = b[i] - m[i] * s;
}

// src[o][c][tap] -> dst[tap][c/2][o][2]  (channel-pair packed, o zero-padded)
__global__ void transpose_w_kernel(const float* __restrict__ src, float* __restrict__ dst,
                                   int O, int Opad, int C, int KK, int total) {
  int i = blockIdx.x * blockDim.x + threadIdx.x;
  if (i >= total) return;
  int j   = i & 1;
  int o   = (i >> 1) % Opad;
  int rem = (i >> 1) / Opad;
  int c2  = rem % (C / 2);
  int t   = rem / (C / 2);
  int c   = 2 * c2 + j;
  dst[i] = (o < O) ? src[(o * C + c) * KK + t] : 0.f;
}

// ------------------------------------------------------- 3x3 conv (implicit GEMM)
// One workgroup: one (b,h) image row (32 pixels) x (blockDim/2) output channels.
// Each wave: 16 O rows x 32 pixels = two 16x16 WMMA accumulators.
template <int COUTPAD>
__global__ void conv3x3_wmma_kernel(const float* __restrict__ src,      // [B][CIN][32][32]
                                    const float* __restrict__ wTp,      // [9][CIN/2][COUTPAD][2]
                                    const float* __restrict__ scale,    // null => 1.0
                                    const float* __restrict__ shift,    // [coutReal]
                                    const float* __restrict__ residual, // null => none
                                    float* __restrict__ dst,            // [B][coutReal][32][32]
                                    int coutReal, int doRelu) {
  __shared__ float Bs[2][32][34];   // [buf][pixel][channel], pitch 34 (even, conflict-free)
  const int tid   = threadIdx.x;
  const int lane  = tid & 31;
  const int wave  = tid >> 5;
  const int khalf = lane >> 4;      // 0: lane holds K0,K1   1: lane holds K2,K3
  const int nl    = lane & 15;
  const int bh = blockIdx.x;
  const int b  = bh >> 5;
  const int h  = bh & 31;
  const int oWave = blockIdx.y * (blockDim.x >> 1) + wave * 16;

  const float* xb = src + (size_t)b * CIN * BHW;

  v8f acc0 = {};
  v8f acc1 = {};
  int pb = 0;

#pragma unroll 1
  for (int t = 0; t < 9; ++t) {
    const int di = t / 3 - 1, dj = t % 3 - 1;
    const int hh = h + di;
    const bool rowOK = (hh >= 0) && (hh < 32);
#pragma unroll 1
    for (int cb = 0; cb < CIN; cb += 32) {
      // Preload this block's 8 A fragments (one b64 each); prefetch next block.
      const float* wTblk = wTp + (size_t)(t * (CIN / 2) + (cb >> 1)) * (COUTPAD * 2) +
                           (oWave + nl) * 2;
      __builtin_prefetch(wTblk + 16 * COUTPAD * 2, 0, 1);
      v2f aReg[8];
#pragma unroll
      for (int s = 0; s < 8; ++s)
        aReg[s] = *(const v2f*)(wTblk + (size_t)(s * 2 + khalf) * (COUTPAD * 2));

      // Stage B tile (pixel-major) into buffer pb.
      for (int idx = tid; idx < 32 * 32; idx += blockDim.x) {
        const int ci = idx >> 5, w = idx & 31;
        const int ww = w + dj;
        float val = 0.f;
        if (rowOK && ww >= 0 && ww < 32)
          val = xb[(size_t)(cb + ci) * BHW + hh * 32 + ww];
        Bs[pb][w][ci] = val;
      }
      __syncthreads();

#pragma unroll
      for (int s = 0; s < 8; ++s) {
        const int k0 = s * 4 + khalf * 2;
        const v2f b0 = *(const v2f*)&Bs[pb][nl][k0];
        const v2f b1 = *(const v2f*)&Bs[pb][nl + 16][k0];
        acc0 = wmma4(aReg[s], b0, acc0);
        acc1 = wmma4(aReg[s], b1, acc1);
      }
      pb ^= 1;
    }
  }

  // Epilogue: BN (+residual) (+ReLU). C/D layout: VGPR r -> M = r + khalf*8.
#pragma unroll
  for (int r = 0; r < 8; ++r) {
    const int o = oWave + khalf * 8 + r;
    if (o >= coutReal) continue;
    const float sc = scale ? scale[o] : 1.0f;
    const float sh = shift[o];
    const size_t base = ((size_t)b * coutReal + o) * BHW + (size_t)h * 32;
    float v0 = acc0[r] * sc + sh;
    float v1 = acc1[r] * sc + sh;
    if (residual) {
      v0 += residual[base + nl];
      v1 += residual[base + nl + 16];
    }
    if (doRelu) { v0 = fmaxf(v0, 0.f); v1 = fmaxf(v1, 0.f); }
    dst[base + nl]      = v0;
    dst[base + nl + 16] = v1;
  }
}

// ------------------------------------------------------- deformable 5x5 conv
// Bilinear sample coefficients are channel-independent: compute (4 addrs + 4
// masked weights) per (tap,pixel) once into LDS, then run 25 per-tap GEMM
// accumulations over C with WMMA. Epilogue: BN -> ReLU -> + out1 -> buf2.
__global__ void deform_wmma_kernel(const float* __restrict__ x,     // [B][CIN][32][32]
                                   const float* __restrict__ off,   // [B][50][32][32]
                                   const float* __restrict__ wTp3,  // [25][CIN/2][256][2]
                                   const float* __restrict__ scale,
                                   const float* __restrict__ shift,
                                   const float* __restrict__ out1,  // add after relu
                                   float* __restrict__ dst) {
  __shared__ int   sA[4][25 * 32];
  __shared__ float sW[4][25 * 32];
  __shared__ float Bs[2][32][34];
  const int tid   = threadIdx.x;
  const int lane  = tid & 31;
  const int wave  = tid >> 5;
  const int khalf = lane >> 4;
  const int nl    = lane & 15;
  const int bh = blockIdx.x;
  const int b  = bh >> 5;
  const int h  = bh & 31;
  const int oWave = blockIdx.y * 128 + wave * 16;

  const float* xb = x + (size_t)b * CIN * BHW;
  const float* ob = off + (size_t)b * 50 * BHW + (size_t)h * 32;

  // Stage 1: per (tap, pixel) bilinear metadata (torchvision DCNv1 semantics)
  for (int e = tid; e < 25 * 32; e += blockDim.x) {
    const int k = e >> 5, w = e & 31;
    const float dy = ob[(size_t)(2 * k) * BHW + w];
    const float dx = ob[(size_t)(2 * k + 1) * BHW + w];
    const float py = (float)(h + k / 5 - 2) + dy;
    const float px = (float)(w + k % 5 - 2) + dx;
    const float fy0 = floorf(py), fx0 = floorf(px);
    const int   y0 = (int)fy0,    x0 = (int)fx0;
    const float ay = py - fy0,    ax = px - fx0;
    const float wy[2] = {1.f - ay, ay};
    const float wx[2] = {1.f - ax, ax};
#pragma unroll
    for (int n = 0; n < 4; ++n) {
      const int yi = y0 + (n >> 1), xi = x0 + (n & 1);
      const bool ok = (yi >= 0) && (yi < 32) && (xi >= 0) && (xi < 32);
      const int yc = min(max(yi, 0), 31), xc = min(max(xi, 0), 31);
      sA[n][e] = yc * 32 + xc;
      sW[n][e] = ok ? wy[n >> 1] * wx[n & 1] : 0.f;
    }
  }
  __syncthreads();

  v8f acc0 = {};
  v8f acc1 = {};
  int pb = 0;
#pragma unroll 1
  for (int k = 0; k < 25; ++k) {
#pragma unroll 1
    for (int cb = 0; cb < CIN; cb += 32) {
      const float* wTblk = wTp3 + (size_t)(k * (CIN / 2) + (cb >> 1)) * (256 * 2) +
                           (oWave + nl) * 2;
      __builtin_prefetch(wTblk + 16 * 256 * 2, 0, 1);
      v2f aReg[8];
#pragma unroll
      for (int s = 0; s < 8; ++s)
        aReg[s] = *(const v2f*)(wTblk + (size_t)(s * 2 + khalf) * (256 * 2));

      for (int idx = tid; idx < 32 * 32; idx += blockDim.x) {
        const int ci = idx >> 5, w = idx & 31;
        const float* xc = xb + (size_t)(cb + ci) * BHW;
        const int e = k * 32 + w;
        Bs[pb][w][ci] = sW[0][e] * xc[sA[0][e]] + sW[1][e] * xc[sA[1][e]] +
                        sW[2][e] * xc[sA[2][e]] + sW[3][e] * xc[sA[3][e]];
      }
      __syncthreads();

#pragma unroll
      for (int s = 0; s < 8; ++s) {
        const int k0 = s * 4 + khalf * 2;
        const v2f b0 = *(const v2f*)&Bs[pb][nl][k0];
        const v2f b1 = *(const v2f*)&Bs[pb][nl + 16][k0];
        acc0 = wmma4(aReg[s], b0, acc0);
        acc1 = wmma4(aReg[s], b1, acc1);
      }
      pb ^= 1;
    }
  }

#pragma unroll
  for (int r = 0; r < 8; ++r) {
    const int o = oWave + khalf * 8 + r;
    const size_t base = ((size_t)b * 256 + o) * BHW + (size_t)h * 32;
    const float sc = scale[o], sh = shift[o];
    float v0 = fmaxf(acc0[r] * sc + sh, 0.f) + out1[base + nl];
    float v1 = fmaxf(acc1[r] * sc + sh, 0.f) + out1[base + nl + 16];
    dst[base + nl]      = v0;
    dst[base + nl + 16] = v1;
  }
}

// ---------------------------------------------------------------- launcher

extern "C" void kernel_launch(void* const* d_in, const int* in_sizes, int n_in,
                              void* d_out, int out_size, void* d_ws, size_t ws_size,
                              hipStream_t stream) {
  const float* x    = (const float*)d_in[0];
  const float* w1   = (const float*)d_in[1];
  const float* woff = (const float*)d_in[2];
  const float* boff = (const float*)d_in[3];
  const float* w3   = (const float*)d_in[4];
  const float* w2   = (const float*)d_in[5];
  const float* g1 = (const float*)d_in[6];
  const float* b1 = (const float*)d_in[7];
  const float* m1 = (const float*)d_in[8];
  const float* v1 = (const float*)d_in[9];
  const float* g3 = (const float*)d_in[10];
  const float* b3 = (const float*)d_in[11];
  const float* m3 = (const float*)d_in[12];
  const float* v3 = (const float*)d_in[13];
  const float* g2 = (const float*)d_in[14];
  const float* b2 = (const float*)d_in[15];
  const float* m2 = (const float*)d_in[16];
  const float* v2 = (const float*)d_in[17];

  // Workspace layout (floats): ~85.5 MB total
  float* ws   = (float*)d_ws;
  float* out1 = ws;                       // 32*256*1024        = 8388608
  float* buf2 = out1 + 8388608;           // 8388608
  float* offb = buf2 + 8388608;           // 32*50*1024         = 1638400
  float* wT1  = offb + 1638400;           // 9*256*256          = 589824
  float* wT2  = wT1 + 589824;             // 589824
  float* wTo  = wT2 + 589824;             // 9*256*64           = 147456
  float* wT3  = wTo + 147456;             // 25*256*256         = 1638400
  float* sc1  = wT3 + 1638400;
  float* sh1  = sc1 + 256;
  float* sc3  = sh1 + 256;
  float* sh3  = sc3 + 256;
  float* sc2  = sh3 + 256;
  float* sh2  = sc2 + 256;

  bnprep_kernel<<<1, 256, 0, stream>>>(g1, b1, m1, v1, sc1, sh1);
  bnprep_kernel<<<1, 256, 0, stream>>>(g3, b3, m3, v3, sc3, sh3);
  bnprep_kernel<<<1, 256, 0, stream>>>(g2, b2, m2, v2, sc2, sh2);

  int tot;
  tot = 9 * 256 * 256;
  transpose_w_kernel<<<(tot + 255) / 256, 256, 0, stream>>>(w1, wT1, 256, 256, 256, 9, tot);
  transpose_w_kernel<<<(tot + 255) / 256, 256, 0, stream>>>(w2, wT2, 256, 256, 256, 9, tot);
  tot = 9 * 256 * 64;
  transpose_w_kernel<<<(tot + 255) / 256, 256, 0, stream>>>(woff, wTo, 50, 64, 256, 9, tot);
  tot = 25 * 256 * 256;
  transpose_w_kernel<<<(tot + 255) / 256, 256, 0, stream>>>(w3, wT3, 256, 256, 256, 25, tot);

  // conv1 + BN1 + ReLU -> out1
  conv3x3_wmma_kernel<256><<<dim3(1024, 2), 256, 0, stream>>>(
      x, wT1, sc1, sh1, nullptr, out1, 256, 1);
  // offset conv + bias -> offb   (O=50 padded to 64, 4-wave blocks)
  conv3x3_wmma_kernel<64><<<dim3(1024, 1), 128, 0, stream>>>(
      x, wTo, nullptr, boff, nullptr, offb, 50, 0);
  // deform conv + BN3 + ReLU + out1 -> buf2
  deform_wmma_kernel<<<dim3(1024, 2), 256, 0, stream>>>(
      x, offb, wT3, sc3, sh3, out1, buf2);
  // conv2 + BN2 + x residual + ReLU -> d_out
  conv3x3_wmma_kernel<256><<<dim3(1024, 2), 256, 0, stream>>>(
      buf2, wT2, sc2, sh2, x, (float*)d_out, 256, 1);
}